// LabelSmoothingLoss_52853867544983
// MI455X (gfx1250) — compile-verified
//
#include <hip/hip_runtime.h>
#include <math.h>

#define THREADS 256
#define TILE    1024          // float4 elements per tile = 16 KB
#define VEC_PER_THREAD (TILE / THREADS)   // 4 float4 (64 B) per thread per tile
#define PIPE    3             // 3-stage ring buffer: 2 tiles in flight

// ---- CDNA5 async global->LDS path (guarded; falls back to sync copy) ----
#if defined(__HIP_DEVICE_COMPILE__) && __has_builtin(__builtin_amdgcn_global_load_async_to_lds_b128)
#define HAVE_ASYNC_LDS 1
typedef int v4i __attribute__((vector_size(16)));
typedef __attribute__((address_space(1))) v4i g_v4i;   // global int4
typedef __attribute__((address_space(3))) v4i l_v4i;   // LDS int4
#define ASYNC_CP_B128(g, l)                                                     \
  __builtin_amdgcn_global_load_async_to_lds_b128((g_v4i*)(g), (l_v4i*)(l), 0, 0)
#if __has_builtin(__builtin_amdgcn_s_wait_asynccnt)
#define WAIT_ASYNC(n) __builtin_amdgcn_s_wait_asynccnt(n)
#else
#define WAIT_ASYNC(n) asm volatile("s_wait_asynccnt %0" ::"i"(n))
#endif
#else
#define WAIT_ASYNC(n)
#endif

// Copy one 16KB tile (TILE float4s) row-tile -> LDS buffer.
__device__ __forceinline__ void cp_tile(const float4* __restrict__ gsrc,
                                        float4* lbuf, int tid) {
#pragma unroll
  for (int k = 0; k < VEC_PER_THREAD; ++k) {
    const int i = k * THREADS + tid;
#ifdef HAVE_ASYNC_LDS
    ASYNC_CP_B128(gsrc + i, lbuf + i);   // async engine writes LDS; ASYNCcnt tracks
#else
    lbuf[i] = gsrc[i];                   // sync fallback (global_load + ds_store)
#endif
  }
}

// Branchless two-phase update for one float4: max, one rescale, 4 exps.
__device__ __forceinline__ void acc_vec4(float4 v, float& m, float& s, float& t) {
  const float mt = fmaxf(fmaxf(v.x, v.y), fmaxf(v.z, v.w));
  const float mn = fmaxf(m, mt);
  s = s * __expf(m - mn);                     // exp(-inf)=0 handles the first call
  s += (__expf(v.x - mn) + __expf(v.y - mn)) +
       (__expf(v.z - mn) + __expf(v.w - mn));
  t += (v.x + v.y) + (v.z + v.w);
  m = mn;
}

__global__ __launch_bounds__(THREADS) void lsloss_rows_kernel(
    const float* __restrict__ logits, const int* __restrict__ target,
    float* __restrict__ row_loss, int V, float conf, float smooth,
    float log_c, float log_s) {
  __shared__ float4 buf[PIPE][TILE];
  __shared__ float sm[THREADS], ss[THREADS], st[THREADS];

  const int row = blockIdx.x;
  const int tid = threadIdx.x;
  const float4* __restrict__ row4 = (const float4*)(logits + (size_t)row * (size_t)V);
  const int nvec      = V >> 2;            // float4s per row (V % 4 == 0)
  const int ntiles    = nvec / TILE;       // full 16KB tiles
  const int tail_base = ntiles * TILE;

  float m = -__builtin_inff(), s = 0.0f, t = 0.0f;

  // Prologue: keep two tiles in flight from the start.
  if (ntiles > 0) cp_tile(row4, &buf[0][0], tid);
  if (ntiles > 1) cp_tile(row4 + TILE, &buf[1][0], tid);

  // Tail float4s loaded directly from global (overlaps with async prologue).
  for (int i = tail_base + tid; i < nvec; i += THREADS) {
    acc_vec4(row4[i], m, s, t);
  }

  // 3-stage ring: copy tile ti+2 while computing tile ti from LDS.
  for (int ti = 0; ti < ntiles; ++ti) {
    const int cur = ti % PIPE;
    if (ti + 2 < ntiles) {
      cp_tile(row4 + (size_t)(ti + 2) * TILE, &buf[(ti + 2) % PIPE][0], tid);
      WAIT_ASYNC(2 * VEC_PER_THREAD);   // tiles ti+1, ti+2 may stay outstanding
    } else if (ti + 1 < ntiles) {
      WAIT_ASYNC(VEC_PER_THREAD);       // tile ti+1 may stay outstanding
    } else {
      WAIT_ASYNC(0);                    // drain for the final tile
    }
    __syncthreads();                    // all waves' tile-ti data visible in LDS

    // Pull this thread's 16 elements into registers (ds_load_b128 x4).
    float4 r0 = buf[cur][0 * THREADS + tid];
    float4 r1 = buf[cur][1 * THREADS + tid];
    float4 r2 = buf[cur][2 * THREADS + tid];
    float4 r3 = buf[cur][3 * THREADS + tid];

    // Tile max (full-rate v_max_f32 tree), one rescale, 16 unconditioned exps.
    const float m01 = fmaxf(fmaxf(r0.x, r0.y), fmaxf(r0.z, r0.w));
    const float m23 = fmaxf(fmaxf(r1.x, r1.y), fmaxf(r1.z, r1.w));
    const float m45 = fmaxf(fmaxf(r2.x, r2.y), fmaxf(r2.z, r2.w));
    const float m67 = fmaxf(fmaxf(r3.x, r3.y), fmaxf(r3.z, r3.w));
    const float mt  = fmaxf(fmaxf(m01, m23), fmaxf(m45, m67));
    const float mn  = fmaxf(m, mt);
    s = s * __expf(m - mn);
    s += ((__expf(r0.x - mn) + __expf(r0.y - mn)) +
          (__expf(r0.z - mn) + __expf(r0.w - mn))) +
         ((__expf(r1.x - mn) + __expf(r1.y - mn)) +
          (__expf(r1.z - mn) + __expf(r1.w - mn)));
    s += ((__expf(r2.x - mn) + __expf(r2.y - mn)) +
          (__expf(r2.z - mn) + __expf(r2.w - mn))) +
         ((__expf(r3.x - mn) + __expf(r3.y - mn)) +
          (__expf(r3.z - mn) + __expf(r3.w - mn)));
    t += ((r0.x + r0.y) + (r0.z + r0.w)) + ((r1.x + r1.y) + (r1.z + r1.w)) +
         ((r2.x + r2.y) + (r2.z + r2.w)) + ((r3.x + r3.y) + (r3.z + r3.w));
    m = mn;

    __syncthreads();                    // buf[cur] free for the copy at ti+1
  }

  // Block reduction: (m, s) merged as logsumexp pairs, t as plain sum.
  sm[tid] = m; ss[tid] = s; st[tid] = t;
  __syncthreads();
  for (int off = THREADS / 2; off > 0; off >>= 1) {
    if (tid < off) {
      float m1 = sm[tid], s1 = ss[tid];
      float m2 = sm[tid + off], s2 = ss[tid + off];
      float M = fmaxf(m1, m2);
      ss[tid] = s1 * __expf(m1 - M) + s2 * __expf(m2 - M);
      sm[tid] = M;
      st[tid] += st[tid + off];
    }
    __syncthreads();
  }

  if (tid == 0) {
    const float M = sm[0], S = ss[0], T = st[0];
    const float logZ = M + logf(S);
    const int   tg   = target[row];
    const float xt   = logits[(size_t)row * (size_t)V + (size_t)tg];
    const float lq_t   = xt - logZ;
    const float sum_lq = T - (float)V * logZ;
    const float per_tok = conf * (log_c - lq_t)
                        + smooth * ((float)(V - 1) * log_s - (sum_lq - lq_t));
    row_loss[row] = (tg != 0) ? per_tok : 0.0f;
  }
}

__global__ __launch_bounds__(256) void lsloss_reduce_kernel(
    const float* __restrict__ row_loss, float* __restrict__ out, int n) {
  __shared__ float sh[256];
  float a = 0.0f;
  for (int i = threadIdx.x; i < n; i += 256) a += row_loss[i];
  sh[threadIdx.x] = a;
  __syncthreads();
  for (int off = 128; off > 0; off >>= 1) {
    if ((int)threadIdx.x < off) sh[threadIdx.x] += sh[threadIdx.x + off];
    __syncthreads();
  }
  if (threadIdx.x == 0) out[0] = sh[0];
}

extern "C" void kernel_launch(void* const* d_in, const int* in_sizes, int n_in,
                              void* d_out, int out_size, void* d_ws, size_t ws_size,
                              hipStream_t stream) {
  const float* logits = (const float*)d_in[0];
  const int*   target = (const int*)d_in[1];
  const int rows = in_sizes[1];
  const int V    = in_sizes[0] / rows;

  const float conf   = 0.7f;                     // 1 - 0.3 label smoothing
  const float smooth = 0.3f / (float)(V - 1);
  const float log_c  = logf(conf);
  const float log_s  = logf(smooth);

  float* row_loss = (float*)d_ws;                // rows * 4 bytes of scratch

  lsloss_rows_kernel<<<rows, THREADS, 0, stream>>>(
      logits, target, row_loss, V, conf, smooth, log_c, log_s);
  lsloss_reduce_kernel<<<1, 256, 0, stream>>>(row_loss, (float*)d_out, rows);
}